// SpectralConv2D_one_35527969472859
// MI455X (gfx1250) — compile-verified
//
#include <hip/hip_runtime.h>
#include <hip/hip_bf16.h>

typedef __attribute__((ext_vector_type(2))) float v2f;
typedef __attribute__((ext_vector_type(8))) float v8f;

#define F_OUT 128
#define KK    9      // 3x3 taps
#define HH    64
#define WW    64
#define CC    128
#define OUT1  62
#define OUT2  62
#define OO    (OUT1 * OUT2)        // 3844
#define NCOL  ((size_t)OO * CC)    // per-filter output stride = 492032

// Branch-free A tap: SM_kernel is (128, 9) row-major; K padded to 12 with zeros.
// Clamp the index so the address is always in-bounds, zero via multiply.
__device__ __forceinline__ float tapA(const float* __restrict__ SM, int m, int k) {
    const int   kc   = (k < KK) ? k : (KK - 1);
    const float mask = (k < KK) ? 1.0f : 0.0f;
    return mask * SM[m * KK + kc];
}

// Branch-free B tap: lam[o,k] * x[cols[o,k], c], zero for padded k.
__device__ __forceinline__ float tapB(const float* __restrict__ x,
                                      const float* __restrict__ Lin,
                                      float lout, int base, int c, int k) {
    const int   kc   = (k < KK) ? k : (KK - 1);
    const float mask = (k < KK) ? 1.0f : 0.0f;
    const int   col  = base + (kc / 3) * WW + (kc % 3);   // always a valid index
    const float lam  = Lin[col] - lout;
    return mask * lam * x[(size_t)col * CC + c];
}

__global__ __launch_bounds__(256)
void spectral_conv2d_wmma_f32(const float* __restrict__ x,     // (HH*WW, CC)
                              const float* __restrict__ SM,    // (F_OUT, KK)
                              const float* __restrict__ Lin,   // (HH*WW)
                              const float* __restrict__ Lout,  // (OO)
                              float* __restrict__ out)         // (F_OUT, OO, CC)
{
    const int o     = blockIdx.x;            // output spatial position, 0..3843
    const int wave  = threadIdx.x >> 5;      // 0..7 -> 16-wide c slice
    const int lane  = threadIdx.x & 31;
    const int lhalf = lane & 15;
    const int khalf = (lane >> 4) << 1;      // lanes 0-15 -> K offset 0; 16-31 -> 2

    const int r    = o / OUT2;
    const int cc   = o - r * OUT2;
    const int base = r * WW + cc;            // top-left input index of this window
    const float lout = Lout[o];

    const int c0 = wave * 16;
    const int c  = c0 + lhalf;               // this lane's output channel column

    // ---- A tiles in registers: 8 M-tiles x 3 K-steps, v2f each ----
    v2f A[8][3];
#pragma unroll
    for (int t = 0; t < 8; ++t) {
        const int m = t * 16 + lhalf;
#pragma unroll
        for (int s = 0; s < 3; ++s) {
            const int k0 = 4 * s + khalf;
            A[t][s].x = tapA(SM, m, k0);
            A[t][s].y = tapA(SM, m, k0 + 1);
        }
    }

    // ---- B tiles: 3 K-steps, v2f each (gather from x, scale by lambda) ----
    v2f B[3];
#pragma unroll
    for (int s = 0; s < 3; ++s) {
        const int k0 = 4 * s + khalf;
        B[s].x = tapB(x, Lin, lout, base, c, k0);
        B[s].y = tapB(x, Lin, lout, base, c, k0 + 1);
    }

    // ---- 24 fp32 WMMAs: D = A x B + D ----
    v8f acc[8] = {};
#pragma unroll
    for (int s = 0; s < 3; ++s) {
#pragma unroll
        for (int t = 0; t < 8; ++t) {
            acc[t] = __builtin_amdgcn_wmma_f32_16x16x4_f32(
                /*neg_a=*/false, A[t][s],
                /*neg_b=*/false, B[s],
                /*c_mod=*/(short)0, acc[t],
                /*reuse_a=*/false, /*reuse_b=*/false);
        }
    }

    // ---- fused ReLU + streaming (non-temporal) store ----
    // D layout: lane holds column c = c0+(lane&15); rows f = 16t + v + (lane>=16 ? 8 : 0)
    const int fhalf = (lane >> 4) * 8;
    float* outp = out + (size_t)o * CC + c;
#pragma unroll
    for (int t = 0; t < 8; ++t) {
#pragma unroll
        for (int v = 0; v < 8; ++v) {
            const int f = t * 16 + v + fhalf;
            const float val = __builtin_fmaxf(acc[t][v], 0.0f);
            __builtin_nontemporal_store(val, outp + (size_t)f * NCOL);
        }
    }
}

extern "C" void kernel_launch(void* const* d_in, const int* in_sizes, int n_in,
                              void* d_out, int out_size, void* d_ws, size_t ws_size,
                              hipStream_t stream) {
    (void)in_sizes; (void)n_in; (void)out_size; (void)d_ws; (void)ws_size;
    const float* x    = (const float*)d_in[0];   // (1, 64, 64, 128) fp32
    const float* SM   = (const float*)d_in[1];   // (128, 9) fp32
    const float* Lin  = (const float*)d_in[2];   // (1, 4096) fp32
    const float* Lout = (const float*)d_in[3];   // (3844, 1) fp32
    float* out        = (float*)d_out;           // (128, 62, 62, 128) fp32

    dim3 grid(OO);    // 3844 blocks, one per output spatial position
    dim3 block(256);  // 8 wave32s, one per 16-wide channel slice
    spectral_conv2d_wmma_f32<<<grid, block, 0, stream>>>(x, SM, Lin, Lout, out);
}